// ResidualBlock_upsample_60902636257544
// MI455X (gfx1250) — compile-verified
//
#include <hip/hip_runtime.h>

#define N_IN_   262144
#define N_OUT_  1048576
#define C_IN_   64
#define C_OUT_  32
#define K3_     8
#define R_SUB_  524288
#define EPS_    1e-4f

// packed-pair LDS region strides (in floats): region = 16(or 8) k-steps * 32 cols * 2,
// +32-dword skew so half=1 lanes hit LDS banks 32..63 (conflict-free ds_load_b64)
#define PADR_   (1024 + 32)   // deconv: 16 steps * 32 * 2 = 1024
#define PADS_   (512 + 32)    // subconv: 8 steps * 32 * 2 = 512

typedef __attribute__((ext_vector_type(2))) float v2f;
typedef __attribute__((ext_vector_type(8))) float v8f;

static __device__ __forceinline__ v8f wmma4(v2f a, v2f b, v8f c) {
    // V_WMMA_F32_16X16X4_F32 : D(16x16,f32) = A(16x4,f32) x B(4x16,f32) + C
    return __builtin_amdgcn_wmma_f32_16x16x4_f32(false, a, false, b, (short)0, c, false, false);
}

static __device__ __forceinline__ v2f ldv2(const float* p) { return *(const v2f*)p; }

// ---------------------------------------------------------------- zero fill
__global__ void zero_kernel(float* __restrict__ p, int n) {
    int i = blockIdx.x * blockDim.x + threadIdx.x;
    int stride = gridDim.x * blockDim.x;
    for (; i < n; i += stride) p[i] = 0.0f;
}

// ------------------------------------------------------ per-channel stats
template <int C>
__global__ __launch_bounds__(256) void stats_kernel(const float* __restrict__ x,
                                                    float* __restrict__ sum,
                                                    float* __restrict__ sumsq,
                                                    int nrows) {
    constexpr int G = 256 / C;
    __shared__ float ls[256];
    __shared__ float lss[256];
    const int c = threadIdx.x % C;
    const int g = threadIdx.x / C;
    float s = 0.0f, ss = 0.0f;
    for (int r = blockIdx.x * G + g; r < nrows; r += gridDim.x * G) {
        float v = x[(size_t)r * C + c];
        s += v;
        ss += v * v;
    }
    ls[threadIdx.x] = s;
    lss[threadIdx.x] = ss;
    __syncthreads();
    if (threadIdx.x < C) {
        float a = 0.0f, b = 0.0f;
#pragma unroll
        for (int i = 0; i < G; ++i) { a += ls[i * C + c]; b += lss[i * C + c]; }
        unsafeAtomicAdd(&sum[c], a);
        unsafeAtomicAdd(&sumsq[c], b);
    }
}

// fold BN into per-channel affine: y = x*scale + bias (relu applied at use site)
__global__ void finalize_kernel(const float* __restrict__ sum, const float* __restrict__ sumsq,
                                const float* __restrict__ gamma, const float* __restrict__ beta,
                                float* __restrict__ scale, float* __restrict__ bias,
                                int C, float invN) {
    int c = threadIdx.x;
    if (c < C) {
        float m = sum[c] * invN;
        float v = fmaxf(sumsq[c] * invN - m * m, 0.0f);
        float s = gamma[c] * rsqrtf(v + EPS_);
        scale[c] = s;
        bias[c]  = beta[c] - m * s;
    }
}

// ------------------------------------------- deconv1 + deconv2 (fused, 64->32)
// 1 wave = one 16-row rulebook tile; all 8 waves of a block share kernel-offset k.
// Weights staged in LDS as interleaved (row, row+1) pairs so B-fragments are a
// single 8-byte ds_load_b64 straight into a WMMA-ready even VGPR pair.
__global__ __launch_bounds__(256) void deconv_kernel(
    const float* __restrict__ x,
    const float* __restrict__ sc1, const float* __restrict__ bi1,
    const float* __restrict__ W1, const float* __restrict__ W2,
    const int* __restrict__ in_idx, const int* __restrict__ out_idx,
    float* __restrict__ y1, float* __restrict__ out) {
    __shared__ float w1p[2 * PADR_];
    __shared__ float w2p[2 * PADR_];

    const int blocksPerK = (N_IN_ / 16) / 8;   // 2048
    const int k    = blockIdx.x / blocksPerK;
    const int bInK = blockIdx.x % blocksPerK;

    const float* w1g = W1 + k * C_IN_ * C_OUT_;
    const float* w2g = W2 + k * C_IN_ * C_OUT_;
    // pack: pair p = row/2; region = p&1 (== half that consumes it); step s = p>>1
    for (int i = threadIdx.x; i < C_IN_ * C_OUT_; i += 256) {
        const int r = i >> 5;          // row 0..63
        const int n = i & 31;          // col 0..31
        const int p = r >> 1;
        const int addr = (p & 1) * PADR_ + (((p >> 1) << 5) + n) * 2 + (r & 1);
        w1p[addr] = w1g[i];
        w2p[addr] = w2g[i];
    }
    __syncthreads();

    const int wave = threadIdx.x >> 5;
    const int lane = threadIdx.x & 31;
    const int half = lane >> 4;     // which half-wave
    const int ln   = lane & 15;     // column / row-in-half
    const int koff = half * 2;      // K offset inside 4-wide K block
    const int r0   = (bInK * 8 + wave) * 16;

    // gather one A row per lane (halves split the K columns of rows r0..r0+15)
    const int grow = in_idx[k * N_IN_ + r0 + ln];
    const float* px = x + (size_t)grow * C_IN_;

    const float* b1base = w1p + half * PADR_ + ln * 2;
    const float* b2base = w2p + half * PADR_ + ln * 2;

    v8f acc00 = {}, acc01 = {}, acc10 = {}, acc11 = {};
#pragma unroll
    for (int s = 0; s < 16; ++s) {
        const int c0 = 4 * s + koff;
        // fused BN+ReLU on gathered A fragment (8-byte aligned float2 loads)
        v2f xv = ldv2(px + c0);
        v2f sv = ldv2(sc1 + c0);
        v2f bv = ldv2(bi1 + c0);
        v2f a;
        a.x = fmaxf(fmaf(xv.x, sv.x, bv.x), 0.0f);
        a.y = fmaxf(fmaf(xv.y, sv.y, bv.y), 0.0f);

        const int o = s << 6;              // (s*32 pairs) * 2 floats
        acc00 = wmma4(a, ldv2(b1base + o),      acc00);   // W1, cols 0..15
        acc01 = wmma4(a, ldv2(b1base + o + 32), acc01);   // W1, cols 16..31
        acc10 = wmma4(a, ldv2(b2base + o),      acc10);   // W2, cols 0..15
        acc11 = wmma4(a, ldv2(b2base + o + 32), acc11);   // W2, cols 16..31
    }

    // scatter-add: lane holds D[m = 8*half + r][ln (+16)]
    const int* oi = out_idx + k * N_IN_ + r0 + half * 8;
#pragma unroll
    for (int r = 0; r < 8; ++r) {
        const int orow = oi[r];
        float* py = y1  + (size_t)orow * C_OUT_;
        float* po = out + (size_t)orow * C_OUT_;
        unsafeAtomicAdd(py + ln,      acc00[r]);
        unsafeAtomicAdd(py + 16 + ln, acc01[r]);
        unsafeAtomicAdd(po + ln,      acc10[r]);
        unsafeAtomicAdd(po + 16 + ln, acc11[r]);
    }
}

// ----------------------------------------------------- subconv (32->32)
__global__ __launch_bounds__(256) void subconv_kernel(
    const float* __restrict__ y1,
    const float* __restrict__ sc2, const float* __restrict__ bi2,
    const float* __restrict__ Ws,
    const int* __restrict__ in_idx, const int* __restrict__ out_idx,
    float* __restrict__ out) {
    __shared__ float wsp[2 * PADS_];

    const int blocksPerK = (R_SUB_ / 16) / 8;  // 4096
    const int k    = blockIdx.x / blocksPerK;
    const int bInK = blockIdx.x % blocksPerK;

    const float* wg = Ws + k * C_OUT_ * C_OUT_;
    for (int i = threadIdx.x; i < C_OUT_ * C_OUT_; i += 256) {
        const int r = i >> 5;          // row 0..31
        const int n = i & 31;
        const int p = r >> 1;
        const int addr = (p & 1) * PADS_ + (((p >> 1) << 5) + n) * 2 + (r & 1);
        wsp[addr] = wg[i];
    }
    __syncthreads();

    const int wave = threadIdx.x >> 5;
    const int lane = threadIdx.x & 31;
    const int half = lane >> 4;
    const int ln   = lane & 15;
    const int koff = half * 2;
    const int r0   = (bInK * 8 + wave) * 16;

    const int grow = in_idx[k * R_SUB_ + r0 + ln];
    const float* py = y1 + (size_t)grow * C_OUT_;

    const float* bbase = wsp + half * PADS_ + ln * 2;

    v8f acc0 = {}, acc1 = {};
#pragma unroll
    for (int s = 0; s < 8; ++s) {
        const int c0 = 4 * s + koff;
        v2f xv = ldv2(py + c0);
        v2f sv = ldv2(sc2 + c0);
        v2f bv = ldv2(bi2 + c0);
        v2f a;
        a.x = fmaxf(fmaf(xv.x, sv.x, bv.x), 0.0f);
        a.y = fmaxf(fmaf(xv.y, sv.y, bv.y), 0.0f);

        const int o = s << 6;
        acc0 = wmma4(a, ldv2(bbase + o),      acc0);
        acc1 = wmma4(a, ldv2(bbase + o + 32), acc1);
    }

    const int* oi = out_idx + k * R_SUB_ + r0 + half * 8;
#pragma unroll
    for (int r = 0; r < 8; ++r) {
        const int orow = oi[r];
        float* po = out + (size_t)orow * C_OUT_;
        unsafeAtomicAdd(po + ln,      acc0[r]);
        unsafeAtomicAdd(po + 16 + ln, acc1[r]);
    }
}

// ---------------------------------------------------------------- launch
extern "C" void kernel_launch(void* const* d_in, const int* in_sizes, int n_in,
                              void* d_out, int out_size, void* d_ws, size_t ws_size,
                              hipStream_t stream) {
    const float* x      = (const float*)d_in[0];
    const float* gamma1 = (const float*)d_in[1];
    const float* beta1  = (const float*)d_in[2];
    const float* gamma2 = (const float*)d_in[3];
    const float* beta2  = (const float*)d_in[4];
    const float* W_d1   = (const float*)d_in[5];
    const float* W_d2   = (const float*)d_in[6];
    const float* W_sub  = (const float*)d_in[7];
    const int* dec_in   = (const int*)d_in[8];
    const int* dec_out  = (const int*)d_in[9];
    const int* sub_in   = (const int*)d_in[10];
    const int* sub_out  = (const int*)d_in[11];
    float* out = (float*)d_out;

    const int yElems = N_OUT_ * C_OUT_;  // 33,554,432 floats (128 MB)
    float* y1   = (float*)d_ws;
    float* stat = (float*)((char*)d_ws + (size_t)yElems * sizeof(float));
    float* sum1 = stat;        // 64
    float* sq1  = stat + 64;   // 64
    float* sc1  = stat + 128;  // 64
    float* bi1  = stat + 192;  // 64
    float* sum2 = stat + 256;  // 32
    float* sq2  = stat + 288;  // 32
    float* sc2  = stat + 320;  // 32
    float* bi2  = stat + 352;  // 32

    // zero accumulators + stats
    zero_kernel<<<2048, 256, 0, stream>>>(y1, yElems);
    zero_kernel<<<2048, 256, 0, stream>>>(out, yElems);
    zero_kernel<<<1, 256, 0, stream>>>(stat, 384);

    // bnr1 -> per-channel affine
    stats_kernel<C_IN_><<<512, 256, 0, stream>>>(x, sum1, sq1, N_IN_);
    finalize_kernel<<<1, 64, 0, stream>>>(sum1, sq1, gamma1, beta1, sc1, bi1,
                                          C_IN_, 1.0f / (float)N_IN_);

    // deconv1 (-> y1 pre-BN) and deconv2 (-> out) fused
    deconv_kernel<<<K3_ * ((N_IN_ / 16) / 8), 256, 0, stream>>>(
        x, sc1, bi1, W_d1, W_d2, dec_in, dec_out, y1, out);

    // bnr2 -> per-channel affine
    stats_kernel<C_OUT_><<<512, 256, 0, stream>>>(y1, sum2, sq2, N_OUT_);
    finalize_kernel<<<1, 32, 0, stream>>>(sum2, sq2, gamma2, beta2, sc2, bi2,
                                          C_OUT_, 1.0f / (float)N_OUT_);

    // subconv accumulates into out (out already holds deconv2 contribution)
    subconv_kernel<<<K3_ * ((R_SUB_ / 16) / 8), 256, 0, stream>>>(
        y1, sc2, bi2, W_sub, sub_in, sub_out, out);
}